// TopKRouter_6219112645446
// MI455X (gfx1250) — compile-verified
//
#include <hip/hip_runtime.h>
#include <hip/hip_bf16.h>
#include <math.h>

typedef __attribute__((ext_vector_type(2))) float v2f;
typedef __attribute__((ext_vector_type(4))) float v4f;
typedef __attribute__((ext_vector_type(8))) float v8f;

#define TOKENS   16384
#define DDIM     2048
#define NEXP     64
#define TOPK     8
#define BLK_TOK  128          // tokens per block (8 waves x 16)
#define KC       32           // K chunk
#define A_STRIDE 36           // floats per sA row (conflict-free: 36*r mod 64 period 16)
#define B_STRIDE 80           // float2 per sB row (row stride = 32 banks offset)

__global__ __launch_bounds__(256)
void topk_router_kernel(const float* __restrict__ hs,
                        const float* __restrict__ cent,
                        float* __restrict__ gate,
                        float* __restrict__ idx_out,
                        float* __restrict__ logits_out)
{
    __shared__ __align__(16) float smem[8192];   // 32 KB, re-used across phases
    float* sA = smem;                                  // [128][36]
    v2f*   sB = (v2f*)(smem + BLK_TOK * A_STRIDE);     // [16][80] pair-interleaved

    const int t    = threadIdx.x;
    const int wave = t >> 5;
    const int lane = t & 31;
    const int half = lane >> 4;       // 0: K-pair kk..kk+1, 1: kk+2..kk+3
    const int l16  = lane & 15;
    const int m0   = blockIdx.x * BLK_TOK;
    const int arow = (wave << 4) + l16;

    v8f acc0 = {}, acc1 = {}, acc2 = {}, acc3 = {};

    for (int k0 = 0; k0 < DDIM; k0 += KC) {
        __syncthreads();
        // ---- stage A tile: 128 tokens x 32 K, coalesced b128 loads ----
        #pragma unroll
        for (int i = 0; i < 4; ++i) {
            int e  = t + (i << 8);
            int r  = e >> 3;
            int c4 = (e & 7) << 2;
            v4f v = *(const v4f*)(hs + (size_t)(m0 + r) * DDIM + k0 + c4);
            *(v4f*)(sA + r * A_STRIDE + c4) = v;
            if (i == 0 && k0 + KC < DDIM)
                __builtin_prefetch(hs + (size_t)(m0 + r) * DDIM + k0 + KC + c4, 0, 1);
        }
        // ---- stage B tile: centroids^T, K-pair interleaved [16][80] ----
        #pragma unroll
        for (int i = 0; i < 4; ++i) {
            int e = t + (i << 8);
            int n = e >> 4;
            int p = e & 15;
            v2f v = *(const v2f*)(cent + (size_t)n * DDIM + k0 + (p << 1));
            sB[p * B_STRIDE + n] = v;
        }
        __syncthreads();
        // ---- 8 K-steps of 4, four 16x16 N-tiles each: 32 WMMAs / chunk ----
        #pragma unroll
        for (int kk = 0; kk < KC; kk += 4) {
            v2f a = *(const v2f*)(sA + arow * A_STRIDE + kk + (half << 1));
            int bp = ((kk >> 1) + half) * B_STRIDE + l16;
            v2f b0 = sB[bp];
            v2f b1 = sB[bp + 16];
            v2f b2 = sB[bp + 32];
            v2f b3 = sB[bp + 48];
            acc0 = __builtin_amdgcn_wmma_f32_16x16x4_f32(false, a, false, b0, (short)0, acc0, false, false);
            acc1 = __builtin_amdgcn_wmma_f32_16x16x4_f32(false, a, false, b1, (short)0, acc1, false, false);
            acc2 = __builtin_amdgcn_wmma_f32_16x16x4_f32(false, a, false, b2, (short)0, acc2, false, false);
            acc3 = __builtin_amdgcn_wmma_f32_16x16x4_f32(false, a, false, b3, (short)0, acc3, false, false);
        }
    }

    // ---- dump accumulators to LDS logits tile [128][64] ----
    __syncthreads();
    float* sL = smem;
    #pragma unroll
    for (int v = 0; v < 8; ++v) {
        int tok = (wave << 4) + v + (half << 3);   // C/D layout: VGPR v -> row v + 8*half
        sL[tok * NEXP +      l16] = acc0[v];
        sL[tok * NEXP + 16 + l16] = acc1[v];
        sL[tok * NEXP + 32 + l16] = acc2[v];
        sL[tok * NEXP + 48 + l16] = acc3[v];
    }
    __syncthreads();

    // ---- coalesced router_logits store ----
    {
        const size_t base = (size_t)m0 * NEXP;
        #pragma unroll
        for (int i = 0; i < 8; ++i) {
            int f = (t + (i << 8)) << 2;
            *(v4f*)(logits_out + base + f) = *(v4f*)(sL + f);
        }
    }

    // ---- per-token softmax prep (read logits into registers first) ----
    float larr[NEXP];
    float mx = -INFINITY, denom = 0.0f;
    if (t < BLK_TOK) {
        #pragma unroll
        for (int n = 0; n < NEXP; ++n) {
            larr[n] = sL[t * NEXP + n];
            mx = fmaxf(mx, larr[n]);
        }
        #pragma unroll
        for (int n = 0; n < NEXP; ++n)
            denom += __expf(larr[n] - mx);
    }
    __syncthreads();

    // ---- reuse smem as dense gate tile; zero cooperatively ----
    float* sG = smem;
    #pragma unroll
    for (int i = 0; i < 32; ++i)
        sG[t + (i << 8)] = 0.0f;
    __syncthreads();

    if (t < BLK_TOK) {
        int   tidx[TOPK];
        float tval[TOPK];
        float wsum = 0.0f;
        #pragma unroll
        for (int j = 0; j < TOPK; ++j) {
            float best = -INFINITY; int bi = 0;
            #pragma unroll
            for (int n = 0; n < NEXP; ++n)
                if (larr[n] > best) { best = larr[n]; bi = n; }   // strict > => lowest index on tie
            larr[bi] = -INFINITY;
            tidx[j] = bi;
            float p = __expf(best - mx) / denom;
            tval[j] = p;
            wsum += p;
        }
        float inv = 1.0f / (wsum + 1e-9f);
        #pragma unroll
        for (int j = 0; j < TOPK; ++j) {
            sG[t * NEXP + tidx[j]] = tval[j] * inv;
            idx_out[(size_t)(m0 + t) * TOPK + j] = (float)tidx[j];
        }
    }
    __syncthreads();

    // ---- coalesced gate_weights store ----
    {
        const size_t base = (size_t)m0 * NEXP;
        #pragma unroll
        for (int i = 0; i < 8; ++i) {
            int f = (t + (i << 8)) << 2;
            *(v4f*)(gate + base + f) = *(v4f*)(sG + f);
        }
    }
}

extern "C" void kernel_launch(void* const* d_in, const int* in_sizes, int n_in,
                              void* d_out, int out_size, void* d_ws, size_t ws_size,
                              hipStream_t stream) {
    (void)in_sizes; (void)n_in; (void)d_ws; (void)ws_size; (void)out_size;
    const float* hs   = (const float*)d_in[0];   // [4,4096,2048] f32
    const float* cent = (const float*)d_in[1];   // [64,2048] f32
    float* gate   = (float*)d_out;                         // [16384,64]
    float* idxo   = gate + (size_t)TOKENS * NEXP;          // [16384,8] (as float)
    float* lout   = idxo + (size_t)TOKENS * TOPK;          // [16384,64]
    topk_router_kernel<<<TOKENS / BLK_TOK, 256, 0, stream>>>(hs, cent, gate, idxo, lout);
}